// KernelScDM_16612933501413
// MI455X (gfx1250) — compile-verified
//
#include <hip/hip_runtime.h>
#include <math.h>

// ---------------------------------------------------------------------------
// Diffusion-map style normalized RBF kernel, N=M=8192, D_IN=16, fp32.
// X/Xref are 512KB each -> L2-resident; recompute the 8192x8192 RBF matrix
// tile-by-tile in 4 fused passes instead of materializing it. Gram part of
// each 16x16 tile = 4x V_WMMA_F32_16X16X4_F32. Final 256MB write is
// non-temporal (streaming). Rowsum passes are templated so the hot loop is
// branch-free, and exp uses raw v_exp_f32 (arg is always <= 0; flush-to-zero
// below 2^-126 is numerically irrelevant here) to avoid the libm range-fixup
// (~5 VALU ops/element saved across four 64M-element passes).
// ---------------------------------------------------------------------------

typedef __attribute__((ext_vector_type(2))) float v2f;
typedef __attribute__((ext_vector_type(8))) float v8f;

#define DIM 16
#define LOG2E 1.4426950408889634f
#define NWAVES 8          // 256-thread blocks = 8 wave32

#if defined(__has_builtin) && __has_builtin(__builtin_amdgcn_exp2f)
#define FAST_EXP2(x) __builtin_amdgcn_exp2f(x)   // raw v_exp_f32
#else
#define FAST_EXP2(x) exp2f(x)
#endif

// Load one K-chunk (4 wide) of a 16-row fp32 operand tile in WMMA A/B layout:
// lanes 0-15 hold K = {4c, 4c+1}, lanes 16-31 hold K = {4c+2, 4c+3}, for
// point index (lane & 15).  (CDNA5 ISA "32-bit A-Matrix 16x4")
__device__ __forceinline__ v2f ld_chunk(const float* __restrict__ base,
                                        int pt, int chunk, int half) {
  const float* p = base + pt * DIM + chunk * 4 + half * 2;
  return v2f{p[0], p[1]};
}

// 16x16 fp32 Gram tile: A rows [row tile] x B rows [col tile], K = 16,
// via 4 chained V_WMMA_F32_16X16X4_F32.
__device__ __forceinline__ v8f gram16x16(const v2f a[4],
                                         const float* __restrict__ B,
                                         int col0, int sub, int half) {
  v8f acc = {};
#pragma unroll
  for (int c = 0; c < 4; ++c) {
    v2f b = ld_chunk(B, col0 + sub, c, half);
    acc = __builtin_amdgcn_wmma_f32_16x16x4_f32(
        /*neg_a=*/false, a[c], /*neg_b=*/false, b,
        /*c_mod=*/(short)0, acc, /*reuse_a=*/false, /*reuse_b=*/false);
  }
  return acc;
}

// -------------------------- tiny scalar kernels ----------------------------

__global__ void params_kernel(const float* __restrict__ log_eps,
                              const float* __restrict__ log_t,
                              float* __restrict__ params) {
  if (threadIdx.x == 0 && blockIdx.x == 0) {
    float eps = log1pf(expf(log_eps[0]));   // softplus
    float t   = log1pf(expf(log_t[0]));     // softplus
    params[0] = 1.0f / (4.0f * eps);        // inv_scale2
    params[1] = t;
  }
}

__global__ void norms_kernel(const float* __restrict__ v,
                             float* __restrict__ out, int n) {
  int i = blockIdx.x * blockDim.x + threadIdx.x;
  if (i < n) {
    float s = 0.0f;
#pragma unroll
    for (int k = 0; k < DIM; ++k) {
      float x = v[i * DIM + k];
      s = fmaf(x, x, s);
    }
    out[i] = s;
  }
}

__global__ void pow_kernel(const float* __restrict__ S, float* __restrict__ D,
                           const float* __restrict__ params, int n) {
  int i = blockIdx.x * blockDim.x + threadIdx.x;
  if (i < n) D[i] = powf(S[i], -params[1]);          // S^(-t)
}

// fac = D * (D*U)^(-1/2)
__global__ void fac_kernel(const float* __restrict__ D,
                           const float* __restrict__ U,
                           float* __restrict__ fac, int n) {
  int i = blockIdx.x * blockDim.x + threadIdx.x;
  if (i < n) {
    float d = D[i];
    fac[i] = d * (1.0f / sqrtf(d * U[i]));
  }
}

// ---------------------- fused RBF row-sum pass (WMMA) ----------------------
// For each row i of A: out0[i] = sum_j w_ij, out1[i] = sum_j w_ij * cw[j],
// with w_ij = exp(-max(|a_i|^2 + |b_j|^2 - 2<a_i,b_j>, 0) * inv_scale2).
// One block per 16-row tile; 8 waves stride over the 512 column tiles.
// Deterministic reduction: shfl butterfly -> LDS -> fixed-order sum.
template <bool HAS_CW, bool OUT0, bool OUT1>
__global__ void __launch_bounds__(32 * NWAVES)
rowsum_kernel(const float* __restrict__ A, const float* __restrict__ an,
              const float* __restrict__ B, const float* __restrict__ bn,
              const float* __restrict__ cw,        // column weight (HAS_CW)
              float* __restrict__ out0,            // unweighted sums (OUT0)
              float* __restrict__ out1,            // weighted sums (OUT1)
              const float* __restrict__ params, int M) {
  __shared__ float l0[NWAVES * 16];
  __shared__ float l1[NWAVES * 16];

  const float nls  = -params[0] * LOG2E;
  const int lane = threadIdx.x & 31;
  const int wave = threadIdx.x >> 5;
  const int half = lane >> 4;
  const int sub  = lane & 15;
  const int row0 = blockIdx.x * 16;

  v2f a[4];
#pragma unroll
  for (int c = 0; c < 4; ++c) a[c] = ld_chunk(A, row0 + sub, c, half);

  float anr[8];
#pragma unroll
  for (int r = 0; r < 8; ++r) anr[r] = an[row0 + r + 8 * half];

  float s0[8], s1[8];
#pragma unroll
  for (int r = 0; r < 8; ++r) { s0[r] = 0.0f; s1[r] = 0.0f; }

  const int ntiles = M / 16;
  for (int tt = wave; tt < ntiles; tt += NWAVES) {
    const int col0 = tt * 16;
    v8f g = gram16x16(a, B, col0, sub, half);
    const float bnn = bn[col0 + sub];
    const float cwv = HAS_CW ? cw[col0 + sub] : 1.0f;
#pragma unroll
    for (int r = 0; r < 8; ++r) {
      float d2 = fmaxf(anr[r] + bnn - 2.0f * g[r], 0.0f);
      float w  = FAST_EXP2(nls * d2);
      if (OUT0) s0[r] += w;
      if (OUT1) s1[r] += w * cwv;
    }
  }

  // reduce across the 16 lanes of each half-wave (cols of the tile)
#pragma unroll
  for (int m = 1; m < 16; m <<= 1) {
#pragma unroll
    for (int r = 0; r < 8; ++r) {
      if (OUT0) s0[r] += __shfl_xor(s0[r], m, 16);
      if (OUT1) s1[r] += __shfl_xor(s1[r], m, 16);
    }
  }
  if (sub == 0) {
#pragma unroll
    for (int r = 0; r < 8; ++r) {
      if (OUT0) l0[wave * 16 + r + 8 * half] = s0[r];
      if (OUT1) l1[wave * 16 + r + 8 * half] = s1[r];
    }
  }
  __syncthreads();

  // fixed-order cross-wave reduction -> deterministic output
  if (threadIdx.x < 16) {
    float t0 = 0.0f, t1 = 0.0f;
#pragma unroll
    for (int w = 0; w < NWAVES; ++w) {
      if (OUT0) t0 += l0[w * 16 + threadIdx.x];
      if (OUT1) t1 += l1[w * 16 + threadIdx.x];
    }
    const int row = row0 + (int)threadIdx.x;
    if (OUT0) out0[row] = t0;
    if (OUT1) out1[row] = t1;
  }
}

// ----------------------------- final pass ----------------------------------
// out[i,j] = rowfac[i] * exp(-d2_ij * inv_scale2) * colfac[j]; 256MB NT store.
__global__ void __launch_bounds__(32 * NWAVES)
final_kernel(const float* __restrict__ X, const float* __restrict__ xn,
             const float* __restrict__ R, const float* __restrict__ rn,
             const float* __restrict__ rowfac, const float* __restrict__ colfac,
             const float* __restrict__ params, float* __restrict__ out, int M) {
  const float nls = -params[0] * LOG2E;
  const int lane = threadIdx.x & 31;
  const int wave = threadIdx.x >> 5;
  const int half = lane >> 4;
  const int sub  = lane & 15;
  const int row0 = blockIdx.x * 16;
  const int col0 = (blockIdx.y * NWAVES + wave) * 16;

  v2f a[4];
#pragma unroll
  for (int c = 0; c < 4; ++c) a[c] = ld_chunk(X, row0 + sub, c, half);

  v8f g = gram16x16(a, R, col0, sub, half);
  const float bnn = rn[col0 + sub];
  const float cf  = colfac[col0 + sub];
#pragma unroll
  for (int r = 0; r < 8; ++r) {
    const int row = row0 + r + 8 * half;
    float d2 = fmaxf(xn[row] + bnn - 2.0f * g[r], 0.0f);
    float v  = rowfac[row] * FAST_EXP2(nls * d2) * cf;
    __builtin_nontemporal_store(v, &out[(size_t)row * (size_t)M + (size_t)(col0 + sub)]);
  }
}

// ------------------------------- driver ------------------------------------

extern "C" void kernel_launch(void* const* d_in, const int* in_sizes, int n_in,
                              void* d_out, int out_size, void* d_ws, size_t ws_size,
                              hipStream_t stream) {
  const float* X       = (const float*)d_in[0];
  const float* Xref    = (const float*)d_in[1];
  const float* log_eps = (const float*)d_in[2];
  const float* log_t   = (const float*)d_in[3];
  float* out = (float*)d_out;

  const int N = in_sizes[0] / DIM;   // 8192
  const int M = in_sizes[1] / DIM;   // 8192

  // workspace carve-out (fp32): ~288KB
  float* ws     = (float*)d_ws;
  float* xn     = ws + 0 * (size_t)M;   // |x_i|^2
  float* rn     = ws + 1 * (size_t)M;   // |ref_j|^2
  float* Sref   = ws + 2 * (size_t)M;   // ref unweighted rowsums, reused as Dx
  float* Uref   = ws + 3 * (size_t)M;   // ref weighted rowsums
  float* Sx     = ws + 4 * (size_t)M;   // cross unweighted rowsums
  float* Ux     = ws + 5 * (size_t)M;   // cross weighted rowsums
  float* Dref   = ws + 6 * (size_t)M;
  float* colfac = ws + 7 * (size_t)M;   // Dref * Dinv1ref
  float* rowfac = ws + 8 * (size_t)M;   // Dx * Dinv1x
  float* params = ws + 9 * (size_t)M;   // [inv_scale2, t]
  (void)ws_size; (void)n_in; (void)out_size;

  params_kernel<<<1, 1, 0, stream>>>(log_eps, log_t, params);
  norms_kernel<<<N / 256, 256, 0, stream>>>(X, xn, N);
  norms_kernel<<<M / 256, 256, 0, stream>>>(Xref, rn, M);

  const dim3 blk(32 * NWAVES);

  // Pass 1: Sref = rowsum(Wr)                       (unweighted only)
  rowsum_kernel<false, true, false><<<dim3(M / 16), blk, 0, stream>>>(
      Xref, rn, Xref, rn, nullptr, Sref, nullptr, params, M);
  // Dref = Sref^(-t)
  pow_kernel<<<M / 256, 256, 0, stream>>>(Sref, Dref, params, M);

  // Pass 2: Uref = rowsum(Wr * Dref[j])             (weighted only)
  rowsum_kernel<true, false, true><<<dim3(M / 16), blk, 0, stream>>>(
      Xref, rn, Xref, rn, Dref, nullptr, Uref, params, M);
  // colfac = Dref * (Dref*Uref)^(-1/2)
  fac_kernel<<<M / 256, 256, 0, stream>>>(Dref, Uref, colfac, M);

  // Pass 3 (dual): Sx = rowsum(W), Ux = rowsum(W * Dref[j])
  rowsum_kernel<true, true, true><<<dim3(N / 16), blk, 0, stream>>>(
      X, xn, Xref, rn, Dref, Sx, Ux, params, M);
  // Dx = Sx^(-t) (reuse Sref slot), rowfac = Dx * (Dx*Ux)^(-1/2)
  pow_kernel<<<N / 256, 256, 0, stream>>>(Sx, Sref, params, N);
  fac_kernel<<<N / 256, 256, 0, stream>>>(Sref, Ux, rowfac, N);

  // Pass 4: out[i,j] = rowfac[i] * w_ij * colfac[j]   (streaming NT store)
  final_kernel<<<dim3(N / 16, (M / 16) / NWAVES), blk, 0, stream>>>(
      X, xn, Xref, rn, rowfac, colfac, params, out, M);
}